// GaussRenderer_63161789055253
// MI455X (gfx1250) — compile-verified
//
#include <hip/hip_runtime.h>
#include <hip/hip_bf16.h>
#include <math.h>
#include <stdint.h>

// ---------------------------------------------------------------------------
// 3D Gaussian splatting forward renderer for MI455X (gfx1250, wave32).
//  - exponent eval:  D(16g x 16p) = coeffs(16x8) x features(8x16) via two
//    chained V_WMMA_F32_16X16X4_F32
//  - data movement:  per-stage 8KB gaussian-record tiles DMA'd into LDS by the
//    Tensor Data Mover (tensor_load_to_lds + s_wait_tensorcnt), double-buffered
// ---------------------------------------------------------------------------

#define NG   4096
#define IMG  256

typedef float v2f __attribute__((ext_vector_type(2)));
typedef float v8f __attribute__((ext_vector_type(8)));
typedef unsigned int u32x4 __attribute__((ext_vector_type(4)));
typedef int i32x8 __attribute__((ext_vector_type(8)));
typedef int i32x4 __attribute__((ext_vector_type(4)));

#if defined(__HIP_DEVICE_COMPILE__) && \
    __has_builtin(__builtin_amdgcn_tensor_load_to_lds) && \
    __has_builtin(__builtin_amdgcn_s_wait_tensorcnt)
#define GS_HAVE_TDM 1
#else
#define GS_HAVE_TDM 0
#endif

static __device__ __forceinline__ v2f mkv2(float a, float b) {
  v2f v; v.x = a; v.y = b; return v;
}

#if GS_HAVE_TDM
// Issue one TDM transfer: 8192 bytes (128 gaussian records) global -> LDS.
// D# per CDNA5 ISA ch.8: 1 row of 1024 x 8B elements.
// This toolchain declares the 6-arg builtin:
//   (u32x4 grp0, i32x8 grp1, i32x4 grp2, i32x4 grp3, i32x8 extra, i32 cpol)
static __device__ __forceinline__ void gs_tdm_issue(const float4* gsrc,
                                                    unsigned lds_addr) {
  const unsigned long long ga = (unsigned long long)(uintptr_t)gsrc;
  // group0: count=1 | lds_addr[63:32] | global_addr[120:64] | type=2 [127:126]
  u32x4 g0 = { 1u,
               lds_addr,
               (unsigned)(ga & 0xffffffffu),
               (unsigned)((ga >> 32) & 0x01ffffffu) | 0x80000000u };
  // group1: data_size=3(8B)@[17:16]; tensor_dim0=1024@[79:48];
  //         tensor_dim1=1@[111:80]; tile_dim0=1024@[127:112];
  //         tile_dim1=1@[143:128]; tensor_dim0_stride=1024@[207:160]
  i32x8 g1 = { 0x00030000, 0x04000000, 0x00010000, 0x04000000,
               0x00000001, 0x00000400, 0x00000000, 0x00000000 };
  i32x4 gz4 = { 0, 0, 0, 0 };
  i32x8 gz8 = { 0, 0, 0, 0, 0, 0, 0, 0 };
  __builtin_amdgcn_tensor_load_to_lds(g0, g1, gz4, gz4, gz8, 0);
}
#endif

// ======================= 1) per-gaussian preprocess ========================
__global__ __launch_bounds__(256) void gs_preprocess(
    const float* __restrict__ means3D, const float* __restrict__ scales,
    const float* __restrict__ rotations, const float* __restrict__ opacity,
    const float* __restrict__ shs, const float* __restrict__ viewm,
    const float* __restrict__ projm, const float* __restrict__ cam,
    const float* __restrict__ fovx_p, const float* __restrict__ fovy_p,
    const float* __restrict__ focx_p, const float* __restrict__ focy_p,
    float4* __restrict__ k0123, float2* __restrict__ k45,
    float4* __restrict__ rgbd, float4* __restrict__ rect,
    unsigned long long* __restrict__ keys)
{
  const int i = blockIdx.x * blockDim.x + threadIdx.x;
  if (i >= NG) return;

  const float m0 = means3D[3*i+0], m1 = means3D[3*i+1], m2 = means3D[3*i+2];

  // pts_o @ viewmatrix  (row-vector convention)
  float v4[4];
#pragma unroll
  for (int c = 0; c < 4; ++c)
    v4[c] = m0*viewm[0*4+c] + m1*viewm[1*4+c] + m2*viewm[2*4+c] + viewm[3*4+c];
  // @ projmatrix
  float h[4];
#pragma unroll
  for (int c = 0; c < 4; ++c)
    h[c] = v4[0]*projm[0*4+c] + v4[1]*projm[1*4+c] + v4[2]*projm[2*4+c] + v4[3]*projm[3*4+c];
  const float pw    = h[3] + 1e-6f;
  const float ndcx  = h[0] / pw;
  const float ndcy  = h[1] / pw;
  const float depth = v4[2];
  const bool  vis   = depth >= 0.2f;

  // ---- SH degree-3 color (unnormalized view dirs, per reference) ----
  const float dx = m0 - cam[0], dy = m1 - cam[1], dz = m2 - cam[2];
  const float xx = dx*dx, yy = dy*dy, zz = dz*dz;
  const float xy = dx*dy, yz = dy*dz, xz = dx*dz;
  float bas[16];
  bas[0]  =  0.28209479177387814f;
  bas[1]  = -0.4886025119029199f * dy;
  bas[2]  =  0.4886025119029199f * dz;
  bas[3]  = -0.4886025119029199f * dx;
  bas[4]  =  1.0925484305920792f * xy;
  bas[5]  = -1.0925484305920792f * yz;
  bas[6]  =  0.31539156525252005f * (2.f*zz - xx - yy);
  bas[7]  = -1.0925484305920792f * xz;
  bas[8]  =  0.5462742152960396f * (xx - yy);
  bas[9]  = -0.5900435899266435f * dy * (3.f*xx - yy);
  bas[10] =  2.890611442640554f  * xy * dz;
  bas[11] = -0.4570457994644658f * dy * (4.f*zz - xx - yy);
  bas[12] =  0.3731763325901154f * dz * (2.f*zz - 3.f*xx - 3.f*yy);
  bas[13] = -0.4570457994644658f * dx * (4.f*zz - xx - yy);
  bas[14] =  1.445305721320277f  * dz * (xx - yy);
  bas[15] = -0.5900435899266435f * dx * (xx - 3.f*yy);
  float col[3];
#pragma unroll
  for (int c = 0; c < 3; ++c) {
    float s = 0.f;
#pragma unroll
    for (int k = 0; k < 16; ++k) s += bas[k] * shs[i*48 + k*3 + c];
    col[c] = fmaxf(s + 0.5f, 0.f);
  }

  // ---- 3D covariance from quaternion + scales ----
  const float q0 = rotations[4*i+0], q1 = rotations[4*i+1];
  const float q2 = rotations[4*i+2], q3 = rotations[4*i+3];
  const float qn = rsqrtf(q0*q0 + q1*q1 + q2*q2 + q3*q3);
  const float r = q0*qn, x = q1*qn, y = q2*qn, z = q3*qn;
  float R[3][3] = {
    {1.f-2.f*(y*y+z*z), 2.f*(x*y-r*z),     2.f*(x*z+r*y)},
    {2.f*(x*y+r*z),     1.f-2.f*(x*x+z*z), 2.f*(y*z-r*x)},
    {2.f*(x*z-r*y),     2.f*(y*z+r*x),     1.f-2.f*(x*x+y*y)}};
  const float sc[3] = {scales[3*i+0], scales[3*i+1], scales[3*i+2]};
  float Lm[3][3];
#pragma unroll
  for (int a = 0; a < 3; ++a)
#pragma unroll
    for (int b = 0; b < 3; ++b) Lm[a][b] = R[a][b] * sc[b];
  float cov[3][3];
#pragma unroll
  for (int a = 0; a < 3; ++a)
#pragma unroll
    for (int b = 0; b < 3; ++b)
      cov[a][b] = Lm[a][0]*Lm[b][0] + Lm[a][1]*Lm[b][1] + Lm[a][2]*Lm[b][2];

  // ---- EWA 2D projection: J @ Wm @ cov3d @ Wm^T @ J^T ----
  const float tan_fx = tanf(fovx_p[0]*0.5f), tan_fy = tanf(fovy_p[0]*0.5f);
  const float fx = focx_p[0], fy = focy_p[0];
  const float tz = v4[2], invz = 1.f / tz;
  const float limx = 1.3f*tan_fx, limy = 1.3f*tan_fy;
  const float txc = fminf(fmaxf(v4[0]*invz, -limx), limx) * tz;
  const float tyc = fminf(fmaxf(v4[1]*invz, -limy), limy) * tz;
  const float J00 = fx*invz, J02 = -txc*fx*invz*invz;
  const float J11 = fy*invz, J12 = -tyc*fy*invz*invz;
  float T0[3], T1[3];                     // T = J(0:2) @ Wm, Wm[k][j]=V[j][k]
#pragma unroll
  for (int j = 0; j < 3; ++j) {
    T0[j] = J00*viewm[j*4+0] + J02*viewm[j*4+2];
    T1[j] = J11*viewm[j*4+1] + J12*viewm[j*4+2];
  }
  float U0[3], U1[3];
#pragma unroll
  for (int j = 0; j < 3; ++j) {
    U0[j] = T0[0]*cov[0][j] + T0[1]*cov[1][j] + T0[2]*cov[2][j];
    U1[j] = T1[0]*cov[0][j] + T1[1]*cov[1][j] + T1[2]*cov[2][j];
  }
  const float ca = U0[0]*T0[0]+U0[1]*T0[1]+U0[2]*T0[2] + 0.3f;
  const float cb = U0[0]*T1[0]+U0[1]*T1[1]+U0[2]*T1[2];
  const float cc = U1[0]*T0[0]+U1[1]*T0[1]+U1[2]*T0[2];
  const float cd = U1[0]*T1[0]+U1[1]*T1[1]+U1[2]*T1[2] + 0.3f;

  const float mx = ((ndcx + 1.f)*(float)IMG - 1.f)*0.5f;
  const float my = ((ndcy + 1.f)*(float)IMG - 1.f)*0.5f;
  const float det = ca*cd - cb*cc;
  const float mid = 0.5f*(ca + cd);
  const float sq  = sqrtf(fmaxf(mid*mid - det, 0.1f));
  const float lam = fmaxf(mid + sq, mid - sq);
  const float radii = 3.f*ceilf(sqrtf(lam));
  float rminx = fminf(fmaxf(mx - radii, 0.f), (float)(IMG-1));
  float rminy = fminf(fmaxf(my - radii, 0.f), (float)(IMG-1));
  float rmaxx = fminf(fmaxf(mx + radii, 0.f), (float)(IMG-1));
  float rmaxy = fminf(fmaxf(my + radii, 0.f), (float)(IMG-1));

  // conic & expanded-quadratic coefficients:
  //   power(p) = k0*px^2 + k1*py^2 + k2*px*py + k3*px + k4*py + k5
  const float inv = 1.f / det;
  const float c00 = cd*inv, c11 = ca*inv, cs = -(cb + cc)*inv;
  float k0 = -0.5f*c00, k1 = -0.5f*c11, k2 = -0.5f*cs;
  float k3 = c00*mx + 0.5f*cs*my;
  float k4 = c11*my + 0.5f*cs*mx;
  float k5 = -0.5f*(c00*mx*mx + c11*my*my + cs*mx*my) + logf(opacity[i]);
  if (!vis) {                      // kill contribution, avoid NaN leakage
    k0 = k1 = k2 = k3 = k4 = 0.f; k5 = -1e30f;
    rminx = rminy = 3e8f; rmaxx = rmaxy = -3e8f;
  }

  k0123[i] = make_float4(k0, k1, k2, k3);
  k45[i]   = make_float2(k4, k5);
  rgbd[i]  = make_float4(col[0], col[1], col[2], depth);
  rect[i]  = make_float4(rminx, rminy, rmaxx, rmaxy);
  const unsigned int bits = vis ? __float_as_uint(depth) : 0x7F800000u;
  keys[i] = ((unsigned long long)bits << 32) | (unsigned int)i;   // stable
}

// ======================= 2) bitonic sort of depth keys =====================
__global__ __launch_bounds__(1024) void gs_sort(unsigned long long* __restrict__ keys)
{
  __shared__ unsigned long long sk[NG];
  const int tid = threadIdx.x;
  for (int i = tid; i < NG; i += 1024) sk[i] = keys[i];
  __syncthreads();
  for (int k = 2; k <= NG; k <<= 1) {
    for (int j = k >> 1; j > 0; j >>= 1) {
      for (int i = tid; i < NG; i += 1024) {
        const int ixj = i ^ j;
        if (ixj > i) {
          const unsigned long long a = sk[i], b = sk[ixj];
          const bool up = (i & k) == 0;
          if ((a > b) == up) { sk[i] = b; sk[ixj] = a; }
        }
      }
      __syncthreads();
    }
  }
  for (int i = tid; i < NG; i += 1024) keys[i] = sk[i];
}

// ================= 3) gather into sorted 64B/gaussian records ==============
// record j (4 x float4): [ k0..k3 | k4,k5,0,0 | r,g,b,depth | rect ]
__global__ __launch_bounds__(256) void gs_gather(
    const unsigned long long* __restrict__ keys,
    const float4* __restrict__ uk, const float2* __restrict__ uk45,
    const float4* __restrict__ ur, const float4* __restrict__ urc,
    float4* __restrict__ rec)
{
  const int j = blockIdx.x * blockDim.x + threadIdx.x;
  if (j >= NG) return;
  const int i = (int)(keys[j] & 0xffffffffULL);
  const float2 kv = uk45[i];
  rec[j*4 + 0] = uk[i];
  rec[j*4 + 1] = make_float4(kv.x, kv.y, 0.f, 0.f);
  rec[j*4 + 2] = ur[i];
  rec[j*4 + 3] = urc[i];
}

// ======================= 4) tiled WMMA compositor ==========================
// Wave handles a 16-pixel row segment. Per 16-gaussian chunk:
//   D(16g x 16p) = A(coeffs 16x8) x B(features 8x16) via 2x wmma_f32_16x16x4.
// D layout: lane l -> pixel (l&15); VGPRs r hold gaussians r+8*(l>=16).
// Stages of 128 gaussians (8KB) are TDM-DMA'd into LDS, double-buffered.
__global__ __launch_bounds__(256) void gs_render(
    const float4* __restrict__ srec, float* __restrict__ out)
{
  __shared__ __align__(16) float4 LREC[2][512];   // 2 x 8KB stage buffers

  const int tile = blockIdx.x >> 5;          // 16 tiles of 64x64
  const int sub  = blockIdx.x & 31;          // 32 sub-blocks (16x8 px each)
  const int ox = (tile & 3) * 64, oy = (tile >> 2) * 64;
  const int wave = threadIdx.x >> 5;
  const int lane = threadIdx.x & 31;
  const bool hi  = lane >= 16;
  const int px = ox + ((sub >> 3) << 4) + (lane & 15);
  const int py = oy + ((sub & 7) << 3) + wave;
  const float fpx = (float)px, fpy = (float)py;

  // B: pixel features (loop invariant).  K rows: [px^2,py^2,px*py,px | py,1,0,0]
  const v2f bf1 = hi ? mkv2(fpx*fpy, fpx) : mkv2(fpx*fpx, fpy*fpy);
  const v2f bf2 = hi ? mkv2(0.f, 0.f)     : mkv2(fpy, 1.f);

  const float tox = (float)ox, toy = (float)oy;
  const float tex = (float)(ox + 63), tey = (float)(oy + 63);
  const int gsel = lane & 15;

  float Trun = 1.f, aR = 0.f, aG = 0.f, aB = 0.f, aD = 0.f, aA = 0.f;

  // ---- prologue: DMA stage 0 into buffer 0 ----
#if GS_HAVE_TDM
  if (threadIdx.x < 32) {
    gs_tdm_issue(srec, (unsigned)(uintptr_t)&LREC[0][0]);
    __builtin_amdgcn_s_wait_tensorcnt(0);
  }
#else
  for (int idx = threadIdx.x; idx < 512; idx += 256) LREC[0][idx] = srec[idx];
#endif
  __syncthreads();

  for (int stage = 0; stage < NG; stage += 128) {
    const int parity = (stage >> 7) & 1;
    const float4* cur = &LREC[parity][0];

    // kick off DMA of the next stage into the alternate buffer
    if (stage + 128 < NG) {
#if GS_HAVE_TDM
      if (threadIdx.x < 32)
        gs_tdm_issue(srec + (stage + 128) * 4,
                     (unsigned)(uintptr_t)&LREC[parity ^ 1][0]);
#else
      for (int idx = threadIdx.x; idx < 512; idx += 256)
        LREC[parity ^ 1][idx] = srec[(stage + 128) * 4 + idx];
#endif
    }

    for (int c0 = 0; c0 < 128; c0 += 16) {
      const int gofs = c0 + (hi ? 8 : 0);
      const int ga = (c0 + gsel) << 2;
      const float4 kk  = cur[ga + 0];
      const float4 kv4 = cur[ga + 1];
      const float4 rc  = cur[ga + 3];
      const bool m = (fminf(rc.z, tex) > fmaxf(rc.x, tox)) &&
                     (fminf(rc.w, tey) > fmaxf(rc.y, toy));
      const float k5m = m ? kv4.y : -1e30f;     // per-tile cull -> alpha = 0

      float pwv[8];
#if __has_builtin(__builtin_amdgcn_wmma_f32_16x16x4_f32)
      const v2f a1 = hi ? mkv2(kk.z, kk.w) : mkv2(kk.x, kk.y);
      const v2f a2 = hi ? mkv2(0.f, 0.f)   : mkv2(kv4.x, k5m);
      v8f cz = {};
      v8f d = __builtin_amdgcn_wmma_f32_16x16x4_f32(false, a1, false, bf1,
                                                    (short)0, cz, false, false);
      d     = __builtin_amdgcn_wmma_f32_16x16x4_f32(false, a2, false, bf2,
                                                    (short)0, d,  false, false);
#pragma unroll
      for (int r2 = 0; r2 < 8; ++r2) pwv[r2] = d[r2];
#else
#pragma unroll
      for (int r2 = 0; r2 < 8; ++r2) {
        const int gb = (gofs + r2) << 2;
        const float4 kk2  = cur[gb + 0];
        const float4 kv2  = cur[gb + 1];
        const float4 rc2  = cur[gb + 3];
        const bool m2 = (fminf(rc2.z, tex) > fmaxf(rc2.x, tox)) &&
                        (fminf(rc2.w, tey) > fmaxf(rc2.y, toy));
        pwv[r2] = kk2.x*fpx*fpx + kk2.y*fpy*fpy + kk2.z*fpx*fpy +
                  kk2.w*fpx + kv2.x*fpy + (m2 ? kv2.y : -1e30f);
      }
#endif
      // sequential blend of this lane-half's 8 gaussians (in order)
      float Tl = 1.f, sr = 0.f, sg = 0.f, sb = 0.f, sd = 0.f, sa = 0.f;
#pragma unroll
      for (int r2 = 0; r2 < 8; ++r2) {
        const float4 col = cur[((gofs + r2) << 2) + 2];
        const float alpha = fminf(__expf(pwv[r2]), 0.99f);
        const float w = alpha * Tl;
        sr += w * col.x; sg += w * col.y; sb += w * col.z;
        sd += w * col.w; sa += w;
        Tl *= (1.f - alpha);
      }
      // merge halves: lanes l & l+16 hold the same pixel (gauss 0-7 / 8-15)
      const float o_sr = __shfl_xor(sr, 16), o_sg = __shfl_xor(sg, 16);
      const float o_sb = __shfl_xor(sb, 16), o_sd = __shfl_xor(sd, 16);
      const float o_sa = __shfl_xor(sa, 16), o_T  = __shfl_xor(Tl, 16);
      const float lr = hi ? o_sr : sr, hr = hi ? sr : o_sr;
      const float lg = hi ? o_sg : sg, hg = hi ? sg : o_sg;
      const float lb = hi ? o_sb : sb, hb = hi ? sb : o_sb;
      const float ld = hi ? o_sd : sd, hd = hi ? sd : o_sd;
      const float la = hi ? o_sa : sa, ha = hi ? sa : o_sa;
      const float lT = hi ? o_T  : Tl, hT = hi ? Tl : o_T;
      aR += Trun * (lr + lT * hr);
      aG += Trun * (lg + lT * hg);
      aB += Trun * (lb + lT * hb);
      aD += Trun * (ld + lT * hd);
      aA += Trun * (la + lT * ha);
      Trun *= lT * hT;
    }

    // next stage's DMA must be complete before anyone reads the other buffer
#if GS_HAVE_TDM
    if (threadIdx.x < 32) __builtin_amdgcn_s_wait_tensorcnt(0);
#endif
    __syncthreads();
  }

  if (!hi) {                                  // low half owns the pixel
    const int o = (py * IMG + px) * 5;
    out[o + 0] = aR + (1.f - aA);             // WHITE_BKGD
    out[o + 1] = aG + (1.f - aA);
    out[o + 2] = aB + (1.f - aA);
    out[o + 3] = aD;
    out[o + 4] = aA;
  }
}

// ============================ launcher =====================================
extern "C" void kernel_launch(void* const* d_in, const int* in_sizes, int n_in,
                              void* d_out, int out_size, void* d_ws, size_t ws_size,
                              hipStream_t stream) {
  (void)in_sizes; (void)n_in; (void)out_size; (void)ws_size;
  const float* means3D   = (const float*)d_in[0];
  const float* scales    = (const float*)d_in[1];
  const float* rotations = (const float*)d_in[2];
  const float* opacity   = (const float*)d_in[3];
  const float* shs       = (const float*)d_in[4];
  const float* viewm     = (const float*)d_in[5];
  const float* projm     = (const float*)d_in[6];
  const float* cam       = (const float*)d_in[7];
  const float* fovx      = (const float*)d_in[8];
  const float* fovy      = (const float*)d_in[9];
  const float* focx      = (const float*)d_in[10];
  const float* focy      = (const float*)d_in[11];

  char* ws = (char*)d_ws;                     // ~512 KB used
  unsigned long long* keys = (unsigned long long*)(ws + 0);       // 32 KB
  float4* uk0123 = (float4*)(ws + 32768);
  float2* uk45   = (float2*)(ws + 98304);
  float4* urgbd  = (float4*)(ws + 131072);
  float4* urect  = (float4*)(ws + 196608);
  float4* srec   = (float4*)(ws + 262144);    // 4096 x 64B sorted records

  gs_preprocess<<<NG/256, 256, 0, stream>>>(means3D, scales, rotations, opacity,
      shs, viewm, projm, cam, fovx, fovy, focx, focy,
      uk0123, uk45, urgbd, urect, keys);
  gs_sort<<<1, 1024, 0, stream>>>(keys);
  gs_gather<<<NG/256, 256, 0, stream>>>(keys, uk0123, uk45, urgbd, urect, srec);
  gs_render<<<512, 256, 0, stream>>>(srec, (float*)d_out);
}